// WassersteinLoss_72722386256326
// MI455X (gfx1250) — compile-verified
//
#include <hip/hip_runtime.h>
#include <math.h>

// ---------------------------------------------------------------------------
// Sinkhorn divergence (geomloss p=2, blur=1, scaling=0.9) on 48x48 grids, B=8.
// softmin(eps,h)_i = -eps*(M + log sum_j exp(-C_ij/eps) * exp(h_j - M))
// => fp32 GEMM  S = K @ E  with K computed on the fly from pixel coords,
//    executed with V_WMMA_F32_16X16X4_F32 (wave32, A=16x4 f32, B=4x16 f32).
// ---------------------------------------------------------------------------

constexpr int NPIX  = 48 * 48;   // 2304
constexpr int WIDTH = 48;
constexpr int NB    = 8;         // batch
constexpr int NCOL  = 32;        // 4 potentials x 8 batches

// workspace layout (in floats)
constexpr size_t O_LA = 0;
constexpr size_t O_LB = O_LA + (size_t)NB * NPIX;
constexpr size_t O_F  = O_LB + (size_t)NB * NPIX;
constexpr size_t O_G  = O_F  + (size_t)NB * NPIX;
constexpr size_t O_A  = O_G  + (size_t)NB * NPIX;
constexpr size_t O_Bp = O_A  + (size_t)NB * NPIX;
constexpr size_t O_PN = O_Bp + (size_t)NB * NPIX;
constexpr size_t O_QN = O_PN + (size_t)NB * NPIX;
constexpr size_t O_PS = O_QN + (size_t)NB * NPIX;
constexpr size_t O_QS = O_PS + NB;
constexpr size_t O_M  = O_QS + NB;                 // 32 column maxes
constexpr size_t O_S  = O_M + NCOL;                // GEMM out [NPIX][32]
constexpr size_t O_E  = O_S + (size_t)NPIX * NCOL; // exp-shifted cols [NPIX][32]
constexpr size_t WS_FLOATS = O_E + (size_t)NPIX * NCOL;  // ~295k floats ~1.2MB
static_assert((O_E % 4) == 0 && (O_S % 4) == 0, "16B alignment for float4");

typedef float v2f __attribute__((ext_vector_type(2)));
typedef float v8f __attribute__((ext_vector_type(8)));

// ------------------------------- init --------------------------------------
// one block per batch: sums, normalized weights, log-weights
__global__ __launch_bounds__(256) void k_init(const float* __restrict__ pred,
                                              const float* __restrict__ target,
                                              float* __restrict__ ws) {
  const int b = blockIdx.x, tid = threadIdx.x;
  const float* p = pred + ((size_t)b * 2 + 1) * NPIX;  // pred[:,1]
  const float* q = target + (size_t)b * NPIX;          // target[:,0]
  __shared__ float red[256];
  float s1 = 0.f, s2 = 0.f;
  for (int j = tid; j < NPIX; j += 256) { s1 += p[j]; s2 += q[j]; }
  red[tid] = s1; __syncthreads();
  for (int o = 128; o > 0; o >>= 1) { if (tid < o) red[tid] += red[tid + o]; __syncthreads(); }
  const float ps = red[0]; __syncthreads();
  red[tid] = s2; __syncthreads();
  for (int o = 128; o > 0; o >>= 1) { if (tid < o) red[tid] += red[tid + o]; __syncthreads(); }
  const float qs = red[0];
  if (tid == 0) { ws[O_PS + b] = ps; ws[O_QS + b] = qs; }
  const float ips = 1.f / fmaxf(ps, 1e-30f);
  const float iqs = 1.f / fmaxf(qs, 1e-30f);
  for (int j = tid; j < NPIX; j += 256) {
    const float pn = p[j] * ips, qn = q[j] * iqs;
    ws[O_PN + (size_t)b * NPIX + j] = pn;
    ws[O_QN + (size_t)b * NPIX + j] = qn;
    ws[O_LA + (size_t)b * NPIX + j] = logf(fmaxf(pn, 1e-38f));
    ws[O_LB + (size_t)b * NPIX + j] = logf(fmaxf(qn, 1e-38f));
  }
}

// --------------------------- build columns ---------------------------------
// column c = pot*8 + b;  pot0: h=lb+g/eps (->f), pot1: h=la+f/eps (->g),
// pot2: h=la+a/eps (->a), pot3: h=lb+b/eps (->b). mode==0: no carry (init).
__global__ __launch_bounds__(256) void k_cols(float* __restrict__ ws, float eps, int use_carry) {
  const int c = blockIdx.x, tid = threadIdx.x;
  const int pot = c >> 3, b = c & 7;
  const size_t boff = (pot == 0 || pot == 3) ? O_LB : O_LA;
  const size_t coff = (pot == 0) ? O_G : (pot == 1) ? O_F : (pot == 2) ? O_A : O_Bp;
  const float* base  = ws + boff + (size_t)b * NPIX;
  const float* carry = ws + coff + (size_t)b * NPIX;
  const float inv_eps = 1.f / eps;
  float hv[9];
  float mx = -3.402823e38f;
  for (int t = 0; t < 9; ++t) {             // 9*256 == 2304
    const int j = t * 256 + tid;
    float h = base[j];
    if (use_carry) h += carry[j] * inv_eps;
    hv[t] = h; mx = fmaxf(mx, h);
  }
  __shared__ float red[256];
  red[tid] = mx; __syncthreads();
  for (int o = 128; o > 0; o >>= 1) { if (tid < o) red[tid] = fmaxf(red[tid], red[tid + o]); __syncthreads(); }
  const float M = red[0];
  if (tid == 0) ws[O_M + c] = M;
  for (int t = 0; t < 9; ++t) {
    const int j = t * 256 + tid;
    ws[O_E + (size_t)j * NCOL + c] = __expf(hv[t] - M);
  }
}

// ------------------------------ WMMA GEMM ----------------------------------
// S[i][c] = sum_j exp(-0.5*d2(i,j)/eps) * E[j][c]
// Block: 256 threads = 8 waves, each wave owns one 16(i)x16(c) output tile.
// blockIdx.x in [0,18): 8 i-tiles each; blockIdx.y in {0,1}: 16-col half.
// B-panel chunks (128 j-rows x 16 cols, 8KB) staged in LDS, shared by 8 waves.
// A-matrix (K tile) computed on the fly: 2 exps/lane/step.
// VGPR layouts per CDNA5 ISA 7.12.2:
//   A 16x4 f32: lane m=l&15 -> M=m; A[0]:K=0|2 (half), A[1]:K=1|3
//   B 4x16 f32: B[0]: K=0|2, N=m;   B[1]: K=1|3, N=m
//   D 16x16 f32: acc[v]: M=v+8*half, N=m
__global__ __launch_bounds__(256) void k_gemm(float* __restrict__ ws, float inv_eps) {
  const float* E = ws + O_E;
  float* S = ws + O_S;
  const int cbase = blockIdx.y * 16;
  const int wave = threadIdx.x >> 5, lane = threadIdx.x & 31;
  const int ibase = (blockIdx.x * 8 + wave) * 16;
  const int half = lane >> 4, m = lane & 15;
  const int i = ibase + m;
  const float xi = (float)(i % WIDTH), yi = (float)(i / WIDTH);
  const float neg_half_inv_eps = -0.5f * inv_eps;

  __shared__ float sE[128 * 16];   // 8 KB chunk of the B-panel
  v8f acc = {};

  for (int j0 = 0; j0 < NPIX; j0 += 128) {
    __syncthreads();
    // cooperative 128x16 float chunk load (2 x float4 per thread)
    for (int u = 0; u < 2; ++u) {
      const int fi = threadIdx.x * 2 + u;      // [0,512)
      const int r = fi >> 2, seg = fi & 3;
      *(float4*)(sE + r * 16 + seg * 4) =
          *(const float4*)(E + (size_t)(j0 + r) * NCOL + cbase + seg * 4);
    }
    __syncthreads();
    if (j0 + 128 < NPIX)  // hint next panel chunk toward L2 (global_prefetch_b8)
      __builtin_prefetch(E + (size_t)(j0 + 128) * NCOL + cbase, 0, 1);

    for (int jr = 0; jr < 128; jr += 4) {
      const int jA = j0 + jr + 2 * half;
      // A operand: K(i, jA), K(i, jA+1)
      const float xj0 = (float)(jA % WIDTH),       yj0 = (float)(jA / WIDTH);
      const float xj1 = (float)((jA + 1) % WIDTH), yj1 = (float)((jA + 1) / WIDTH);
      const float dx0 = xi - xj0, dy0 = yi - yj0;
      const float dx1 = xi - xj1, dy1 = yi - yj1;
      v2f Av;
      Av.x = __expf((dx0 * dx0 + dy0 * dy0) * neg_half_inv_eps);
      Av.y = __expf((dx1 * dx1 + dy1 * dy1) * neg_half_inv_eps);
      v2f Bv;
      Bv.x = sE[(jr + 2 * half) * 16 + m];
      Bv.y = sE[(jr + 2 * half + 1) * 16 + m];
      acc = __builtin_amdgcn_wmma_f32_16x16x4_f32(
          /*neg_a=*/false, Av, /*neg_b=*/false, Bv,
          /*c_mod=*/(short)0, acc, /*reuse_a=*/false, /*reuse_b=*/false);
    }
  }
  for (int v = 0; v < 8; ++v)
    S[(size_t)(ibase + v + 8 * half) * NCOL + cbase + m] = acc[v];
}

// ------------------------------ update -------------------------------------
// pot-target: pot0->f, pot1->g, pot2->a, pot3->b
// avg==0: assign (init & final extrapolation); avg==1: 0.5*(old+new) (scan)
__global__ __launch_bounds__(256) void k_upd(float* __restrict__ ws, float eps, int avg) {
  const int flat = blockIdx.x * 256 + threadIdx.x;  // < NPIX*32, flat = i*32+c
  const int i = flat >> 5, c = flat & 31;
  const int pot = c >> 3, b = c & 7;
  const float val = -eps * (ws[O_M + c] + logf(fmaxf(ws[O_S + flat], 1e-38f)));
  const size_t toff = (pot == 0) ? O_F : (pot == 1) ? O_G : (pot == 2) ? O_A : O_Bp;
  float* t = ws + toff + (size_t)b * NPIX + i;
  *t = avg ? 0.5f * (*t + val) : val;
}

// ------------------------------- loss --------------------------------------
__global__ __launch_bounds__(256) void k_loss(const float* __restrict__ ws,
                                              float* __restrict__ out) {
  __shared__ float red[256];
  const int tid = threadIdx.x;
  float total = 0.f;
  const float diag = sqrtf((float)(WIDTH * WIDTH * 2));
  for (int b = 0; b < NB; ++b) {
    float acc = 0.f;
    const size_t o = (size_t)b * NPIX;
    for (int j = tid; j < NPIX; j += 256) {
      acc += ws[O_PN + o + j] * (ws[O_F + o + j] - ws[O_A + o + j])
           + ws[O_QN + o + j] * (ws[O_G + o + j] - ws[O_Bp + o + j]);
    }
    red[tid] = acc; __syncthreads();
    for (int of = 128; of > 0; of >>= 1) { if (tid < of) red[tid] += red[tid + of]; __syncthreads(); }
    if (tid == 0) {
      const float dist = red[0];
      const float w2 = sqrtf(fmaxf(2.f * dist, 1e-12f));
      const float qsv = ws[O_QS + b], psv = ws[O_PS + b];
      total += (qsv < 1e-8f) ? 0.f : ((psv < 1e-8f) ? 1.f : (w2 / diag));
    }
    __syncthreads();
  }
  if (tid == 0) out[0] = total / (float)NB;
}

// ------------------------------- host --------------------------------------
extern "C" void kernel_launch(void* const* d_in, const int* in_sizes, int n_in,
                              void* d_out, int out_size, void* d_ws, size_t ws_size,
                              hipStream_t stream) {
  const float* pred   = (const float*)d_in[0];
  const float* target = (const float*)d_in[1];
  float* ws  = (float*)d_ws;          // needs >= WS_FLOATS*4 ~ 1.2 MB
  float* out = (float*)d_out;

  // eps schedule (matches numpy arange in reference; deterministic)
  const double diam = sqrt(47.0 * 47.0 + 47.0 * 47.0);
  const double log_start = 2.0 * log(diam);
  const double step = 2.0 * log(0.9);
  float eps[96]; int ne = 0;
  eps[ne++] = (float)(diam * diam);
  for (double v = log_start; v > 0.0; v += step) eps[ne++] = (float)exp(v);
  eps[ne++] = 1.0f;                   // blur^2

  const dim3 gemm_grid(NPIX / (16 * 8), 2);   // 18 x 2 blocks, 8 waves each
  const int upd_blocks = (NPIX * NCOL) / 256; // 288

  k_init<<<8, 256, 0, stream>>>(pred, target, ws);

  // init quad: f=softmin(eps0,lb), g=a=softmin(eps0,la), b=softmin(eps0,lb)
  k_cols<<<NCOL, 256, 0, stream>>>(ws, eps[0], /*use_carry=*/0);
  k_gemm<<<gemm_grid, 256, 0, stream>>>(ws, 1.f / eps[0]);
  k_upd<<<upd_blocks, 256, 0, stream>>>(ws, eps[0], /*avg=*/0);

  // scan over full eps schedule with symmetric averaged updates
  for (int t = 0; t < ne; ++t) {
    k_cols<<<NCOL, 256, 0, stream>>>(ws, eps[t], 1);
    k_gemm<<<gemm_grid, 256, 0, stream>>>(ws, 1.f / eps[t]);
    k_upd<<<upd_blocks, 256, 0, stream>>>(ws, eps[t], /*avg=*/1);
  }

  // final extrapolation at eps = blur^2 (assign)
  const float epsf = eps[ne - 1];
  k_cols<<<NCOL, 256, 0, stream>>>(ws, epsf, 1);
  k_gemm<<<gemm_grid, 256, 0, stream>>>(ws, 1.f / epsf);
  k_upd<<<upd_blocks, 256, 0, stream>>>(ws, epsf, /*avg=*/0);

  k_loss<<<1, 256, 0, stream>>>(ws, out);
}